// I2BGNN_27977416966472
// MI455X (gfx1250) — compile-verified
//
#include <hip/hip_runtime.h>

#define D 128
#define BR 64            // rows per GEMM block
#define LDA 132          // LDS stride for A tile (bank step 4 -> conflict-free column reads)
#define LDW 136          // LDS stride for W chunk (bank step 8 -> conflict-free row reads)
#define KCHUNK 32
#define BN_EPS 1e-5f

typedef __attribute__((ext_vector_type(2))) float v2f;
typedef __attribute__((ext_vector_type(8))) float v8f;
typedef __attribute__((__vector_size__(4 * sizeof(int)))) int v4i;

// ---- CDNA5 async global->LDS staging (ASYNCcnt path), with safe fallback ----
#if defined(__HIP_DEVICE_COMPILE__) && __has_builtin(__builtin_amdgcn_global_load_async_to_lds_b128)
#define ASYNC_STAGE 1
#else
#define ASYNC_STAGE 0
#endif

#if ASYNC_STAGE
#define GPTR(p) ((v4i*)(p))   // builtin expects generic int4* (global source)
#define LPTR(p) ((v4i*)(p))   // builtin expects generic int4* (LDS destination)
#if __has_builtin(__builtin_amdgcn_s_wait_asynccnt)
#define WAIT_ASYNC() __builtin_amdgcn_s_wait_asynccnt(0)
#else
#define WAIT_ASYNC() asm volatile("s_wait_asynccnt 0x0" ::: "memory")
#endif
#endif

// ---------------- degree / normalization ----------------
__global__ void k_zero_deg(unsigned* deg, int n) {
    int i = blockIdx.x * blockDim.x + threadIdx.x;
    if (i < n) deg[i] = 0u;
}

__global__ void k_count_deg(const int* __restrict__ dst, unsigned* __restrict__ deg, int e) {
    int i = blockIdx.x * blockDim.x + threadIdx.x;
    if (i < e) atomicAdd(&deg[dst[i]], 1u);
}

__global__ void k_dinv(const unsigned* __restrict__ deg, float* __restrict__ dinv, int n) {
    int i = blockIdx.x * blockDim.x + threadIdx.x;
    if (i < n) dinv[i] = rsqrtf((float)deg[i] + 1.0f);   // deg+1 >= 1, never zero
}

// ---------------- fp32 WMMA GEMM: out[N x 128] = A[N x 128] @ W[128 x 128] ----------------
// Block = 256 threads = 8 waves, computes 64 rows x 128 cols.
// Wave (rs = w&3, ch = w>>2): rows [16*rs,16*rs+16), cols [64*ch,64*ch+64) -> 4 accumulators.
template <bool BIAS, bool RELU>
__global__ __launch_bounds__(256) void k_gemm128(const float* __restrict__ A,
                                                 const float* __restrict__ Wm,
                                                 const float* __restrict__ bias,
                                                 float* __restrict__ out, int n) {
    __shared__ float lA[BR * LDA];        // 33792 B
    __shared__ float lW[KCHUNK * LDW];    // 17408 B
    const int tid = threadIdx.x;
    const int rowBase = blockIdx.x * BR;

    // ---- stage A strip (64 x 128), B128 granules (rows are 16B-aligned in LDS) ----
    for (int i = tid; i < BR * (D / 4); i += 256) {   // 2048 float4, 8 per thread
        const int r = i >> 5;
        const int c4 = (i & 31) << 2;
        const int gr = rowBase + r;
        if (gr < n) {
#if ASYNC_STAGE
            __builtin_amdgcn_global_load_async_to_lds_b128(
                GPTR(A + (size_t)gr * D + c4), LPTR(&lA[r * LDA + c4]), 0, 0);
#else
            const float4 v = *(const float4*)(A + (size_t)gr * D + c4);
            *(float4*)&lA[r * LDA + c4] = v;
#endif
        } else {
            lA[r * LDA + c4 + 0] = 0.0f; lA[r * LDA + c4 + 1] = 0.0f;
            lA[r * LDA + c4 + 2] = 0.0f; lA[r * LDA + c4 + 3] = 0.0f;
        }
    }

    const int lane = tid & 31;
    const int wave = tid >> 5;
    const int half = lane >> 4;          // 0: lanes 0-15, 1: lanes 16-31
    const int l    = lane & 15;
    const int rs   = wave & 3;           // row strip
    const int ch   = wave >> 2;          // column half
    const int rowOff = rs * 16 + l;      // this lane's A row within the block

    v8f acc[4];
#pragma unroll
    for (int t = 0; t < 4; ++t) acc[t] = (v8f){0.f,0.f,0.f,0.f,0.f,0.f,0.f,0.f};

    for (int kbase = 0; kbase < D; kbase += KCHUNK) {
        // ---- stage W chunk [kbase, kbase+32) x 128 ----
        for (int i = tid; i < KCHUNK * (D / 4); i += 256) {   // 1024 float4, 4 per thread
            const int r = i >> 5;
            const int c4 = (i & 31) << 2;
#if ASYNC_STAGE
            __builtin_amdgcn_global_load_async_to_lds_b128(
                GPTR(Wm + (size_t)(kbase + r) * D + c4), LPTR(&lW[r * LDW + c4]), 0, 0);
#else
            const float4 v = *(const float4*)(Wm + (size_t)(kbase + r) * D + c4);
            *(float4*)&lW[r * LDW + c4] = v;
#endif
        }
#if ASYNC_STAGE
        WAIT_ASYNC();                    // drain this wave's async copies (A + W chunk)
#endif
        __syncthreads();

#pragma unroll
        for (int kk = 0; kk < KCHUNK; kk += 4) {
            // A frag 16x4 f32: v0 = K=k+2h, v1 = K=k+2h+1 for lane half h
            // B frag 4x16 f32: v0 = row k+2h, v1 = row k+2h+1 across lanes (N = l)
            const int ka = kk + 2 * half;
            v2f a;
            a.x = lA[rowOff * LDA + kbase + ka];
            a.y = lA[rowOff * LDA + kbase + ka + 1];
#pragma unroll
            for (int t = 0; t < 4; ++t) {
                const int col = ch * 64 + t * 16 + l;
                v2f b;
                b.x = lW[ka * LDW + col];
                b.y = lW[(ka + 1) * LDW + col];
                acc[t] = __builtin_amdgcn_wmma_f32_16x16x4_f32(
                    false, a, false, b, (short)0, acc[t], false, false);
            }
        }
        __syncthreads();
    }

    // C layout: VGPR r holds (M = r + 8*half, N = l)
#pragma unroll
    for (int t = 0; t < 4; ++t) {
        const int col = ch * 64 + t * 16 + l;
        const float bv = BIAS ? bias[col] : 0.0f;
#pragma unroll
        for (int r = 0; r < 8; ++r) {
            const int gr = rowBase + rs * 16 + r + 8 * half;
            if (gr < n) {
                float v = acc[t][r] + bv;
                if (RELU) v = fmaxf(v, 0.0f);
                out[(size_t)gr * D + col] = v;
            }
        }
    }
}

// ---------------- GCN scatter pieces ----------------
// agg[n, 4d..4d+3] = h * dinv[n]^2 + bias   (self-loop term + bias, initializes accumulator)
__global__ void k_self_init(const float* __restrict__ h, const float* __restrict__ dinv,
                            const float* __restrict__ bias, float* __restrict__ agg,
                            long long total4) {
    long long i = (long long)blockIdx.x * blockDim.x + threadIdx.x;
    if (i < total4) {
        const int nidx = (int)(i >> 5);
        const int c4 = (int)(i & 31) << 2;
        const float di = dinv[nidx];
        const float w = di * di;
        const float4 hv = ((const float4*)h)[i];
        const float4 bv = *(const float4*)(bias + c4);
        float4 o;
        o.x = hv.x * w + bv.x; o.y = hv.y * w + bv.y;
        o.z = hv.z * w + bv.z; o.w = hv.w * w + bv.w;
        ((float4*)agg)[i] = o;
    }
}

// agg[dst, :] += h[src, :] * dinv[src]*dinv[dst]   (one wave per edge, float4 per lane)
__global__ void k_aggregate(const int* __restrict__ src, const int* __restrict__ dst,
                            const float* __restrict__ dinv, const float* __restrict__ h,
                            float* __restrict__ agg, int e) {
    long long t = (long long)blockIdx.x * blockDim.x + threadIdx.x;
    const int eidx = (int)(t >> 5);
    const int q = (int)(t & 31);
    if (eidx < e) {
        const int s = src[eidx], dd = dst[eidx];
        const float w = dinv[s] * dinv[dd];
        const float4 hv = ((const float4*)(h + (size_t)s * D))[q];
        float* ap = agg + (size_t)dd * D + (q << 2);
        atomicAdd(ap + 0, hv.x * w);
        atomicAdd(ap + 1, hv.y * w);
        atomicAdd(ap + 2, hv.z * w);
        atomicAdd(ap + 3, hv.w * w);
    }
}

// ---------------- fused ReLU + BatchNorm (training-mode, biased var) ----------------
// one block per feature; stats[f] = mu, stats[128+f] = rsqrt(var + eps)
__global__ __launch_bounds__(256) void k_bn_stats(const float* __restrict__ x,
                                                  float* __restrict__ stats, int n) {
    __shared__ float s0[256], s1[256];
    const int f = blockIdx.x;
    const int t = threadIdx.x;
    float sum = 0.0f, sq = 0.0f;
    for (int i = t; i < n; i += 256) {
        const float v = fmaxf(x[(size_t)i * D + f], 0.0f);   // ReLU precedes BN
        sum += v;
        sq += v * v;
    }
    s0[t] = sum; s1[t] = sq;
    __syncthreads();
    for (int s = 128; s > 0; s >>= 1) {
        if (t < s) { s0[t] += s0[t + s]; s1[t] += s1[t + s]; }
        __syncthreads();
    }
    if (t == 0) {
        const float mu = s0[0] / (float)n;
        const float var = s1[0] / (float)n - mu * mu;
        stats[f] = mu;
        stats[D + f] = rsqrtf(var + BN_EPS);
    }
}

__global__ void k_bn_apply(const float* __restrict__ x, const float* __restrict__ stats,
                           const float* __restrict__ gamma, const float* __restrict__ beta,
                           float* __restrict__ out, long long total4) {
    long long i = (long long)blockIdx.x * blockDim.x + threadIdx.x;
    if (i < total4) {
        const int c4 = (int)(i & 31) << 2;
        const float4 xv = ((const float4*)x)[i];
        const float4 mu = *(const float4*)(stats + c4);
        const float4 rs = *(const float4*)(stats + D + c4);
        const float4 g  = *(const float4*)(gamma + c4);
        const float4 bt = *(const float4*)(beta + c4);
        float4 o;
        o.x = (fmaxf(xv.x, 0.f) - mu.x) * rs.x * g.x + bt.x;
        o.y = (fmaxf(xv.y, 0.f) - mu.y) * rs.y * g.y + bt.y;
        o.z = (fmaxf(xv.z, 0.f) - mu.z) * rs.z * g.z + bt.z;
        o.w = (fmaxf(xv.w, 0.f) - mu.w) * rs.w * g.w + bt.w;
        ((float4*)out)[i] = o;
    }
}

// ---------------- head: logits = h @ lin2_w + b ; log_softmax ----------------
__global__ __launch_bounds__(256) void k_head(const float* __restrict__ h,
                                              const float* __restrict__ w,
                                              const float* __restrict__ b,
                                              float* __restrict__ out, int n) {
    __shared__ float lw[D * 10];
    __shared__ float lb[10];
    for (int i = threadIdx.x; i < D * 10; i += 256) lw[i] = w[i];
    if (threadIdx.x < 10) lb[threadIdx.x] = b[threadIdx.x];
    __syncthreads();

    const int node = blockIdx.x * blockDim.x + threadIdx.x;
    if (node >= n) return;

    float logit[10];
#pragma unroll
    for (int c = 0; c < 10; ++c) logit[c] = lb[c];
    const float* row = h + (size_t)node * D;
    for (int k = 0; k < D; ++k) {
        const float xv = row[k];
#pragma unroll
        for (int c = 0; c < 10; ++c) logit[c] = fmaf(xv, lw[k * 10 + c], logit[c]);
    }
    float m = logit[0];
#pragma unroll
    for (int c = 1; c < 10; ++c) m = fmaxf(m, logit[c]);
    float s = 0.0f;
#pragma unroll
    for (int c = 0; c < 10; ++c) s += expf(logit[c] - m);
    const float lse = m + logf(s);
#pragma unroll
    for (int c = 0; c < 10; ++c) out[(size_t)node * 10 + c] = logit[c] - lse;
}

// ---------------- launcher ----------------
extern "C" void kernel_launch(void* const* d_in, const int* in_sizes, int n_in,
                              void* d_out, int out_size, void* d_ws, size_t ws_size,
                              hipStream_t stream) {
    const float* x      = (const float*)d_in[0];
    const int*   ei     = (const int*)d_in[1];
    const float* W1     = (const float*)d_in[2];
    const float* b1     = (const float*)d_in[3];
    const float* gamma1 = (const float*)d_in[4];
    const float* beta1  = (const float*)d_in[5];
    const float* W2     = (const float*)d_in[6];
    const float* b2     = (const float*)d_in[7];
    const float* gamma2 = (const float*)d_in[8];
    const float* beta2  = (const float*)d_in[9];
    const float* l1w    = (const float*)d_in[10];
    const float* l1b    = (const float*)d_in[11];
    const float* l2w    = (const float*)d_in[12];
    const float* l2b    = (const float*)d_in[13];

    const int N = in_sizes[0] / D;
    const int E = in_sizes[1] / 2;
    const int* src = ei;
    const int* dst = ei + E;

    // workspace layout (floats): bufA[N*D] | bufB[N*D] | dinv[N] | deg[N] | stats[256]
    float* ws = (float*)d_ws;
    const size_t ND = (size_t)N * D;
    float*    bufA  = ws;
    float*    bufB  = ws + ND;
    float*    dinv  = ws + 2 * ND;
    unsigned* deg   = (unsigned*)(dinv + N);
    float*    stats = (float*)(deg + N);

    const long long total4  = (long long)N * (D / 4);      // float4 elements
    const long long ethread = (long long)E * 32;           // one wave per edge
    const int nb  = (N + 255) / 256;
    const int eb  = (E + 255) / 256;
    const int tb4 = (int)((total4 + 255) / 256);
    const int ab  = (int)((ethread + 255) / 256);
    const int gb  = (N + BR - 1) / BR;

    // degrees -> dinv (shared by both conv layers)
    k_zero_deg<<<nb, 256, 0, stream>>>(deg, N);
    k_count_deg<<<eb, 256, 0, stream>>>(dst, deg, E);
    k_dinv<<<nb, 256, 0, stream>>>(deg, dinv, N);

    // ---- layer 1: GCNConv -> ReLU -> BN ----
    k_gemm128<false, false><<<gb, 256, 0, stream>>>(x, W1, nullptr, bufA, N);
    k_self_init<<<tb4, 256, 0, stream>>>(bufA, dinv, b1, bufB, total4);
    k_aggregate<<<ab, 256, 0, stream>>>(src, dst, dinv, bufA, bufB, E);
    k_bn_stats<<<D, 256, 0, stream>>>(bufB, stats, N);
    k_bn_apply<<<tb4, 256, 0, stream>>>(bufB, stats, gamma1, beta1, bufA, total4);

    // ---- layer 2 ----
    k_gemm128<false, false><<<gb, 256, 0, stream>>>(bufA, W2, nullptr, bufB, N);
    k_self_init<<<tb4, 256, 0, stream>>>(bufB, dinv, b2, bufA, total4);
    k_aggregate<<<ab, 256, 0, stream>>>(src, dst, dinv, bufB, bufA, E);
    k_bn_stats<<<D, 256, 0, stream>>>(bufA, stats, N);
    k_bn_apply<<<tb4, 256, 0, stream>>>(bufA, stats, gamma2, beta2, bufB, total4);

    // ---- MLP head ----
    k_gemm128<true, true><<<gb, 256, 0, stream>>>(bufB, l1w, l1b, bufA, N);
    k_head<<<nb, 256, 0, stream>>>(bufA, l2w, l2b, (float*)d_out, N);
}